// ProtoNet_4698694222630
// MI455X (gfx1250) — compile-verified
//
#include <hip/hip_runtime.h>

typedef __attribute__((ext_vector_type(2))) float v2f;
typedef __attribute__((ext_vector_type(4))) float v4f;
typedef __attribute__((ext_vector_type(8))) float v8f;

#define NB 128      // query batch (rows of output)
#define NC 64       // number of classes (cols of output)
#define ND 16000    // flattened feature dim 640*5*5

// ---------------------------------------------------------------------------
// Kernel 1: prototypes[c][d] = (1/128) * sum_b support[c][b][d]
// Pure HBM streaming: 524 MB in, 4 MB out. 128-bit non-temporal loads.
// ---------------------------------------------------------------------------
__global__ void __launch_bounds__(256)
proto_mean_kernel(const float* __restrict__ support, float* __restrict__ proto) {
    const int c  = blockIdx.y;
    const int d4 = (blockIdx.x * blockDim.x + threadIdx.x) * 4;
    if (d4 >= ND) return;
    const v4f* base =
        reinterpret_cast<const v4f*>(support + (size_t)c * NB * ND + d4);
    v4f acc = {0.f, 0.f, 0.f, 0.f};
    #pragma unroll 4
    for (int b = 0; b < NB; ++b) {
        v4f v = __builtin_nontemporal_load(&base[(size_t)b * (ND / 4)]);
        acc += v;
    }
    acc *= (1.0f / (float)NB);
    *reinterpret_cast<v4f*>(proto + (size_t)c * ND + d4) = acc;
}

// ---------------------------------------------------------------------------
// Kernel 2: squared L2 norms of the 128 query rows and 64 prototype rows.
// One 256-thread block per row.
// ---------------------------------------------------------------------------
__global__ void __launch_bounds__(256)
row_sumsq_kernel(const float* __restrict__ query, const float* __restrict__ proto,
                 float* __restrict__ xsq, float* __restrict__ ysq) {
    const int r = blockIdx.x;
    const float* row = (r < NB) ? (query + (size_t)r * ND)
                                : (proto + (size_t)(r - NB) * ND);
    const v4f* row4 = reinterpret_cast<const v4f*>(row);
    float acc = 0.f;
    for (int i = threadIdx.x; i < ND / 4; i += blockDim.x) {
        v4f v = row4[i];
        acc += v.x * v.x + v.y * v.y + v.z * v.z + v.w * v.w;
    }
    #pragma unroll
    for (int off = 16; off > 0; off >>= 1)
        acc += __shfl_xor(acc, off, 32);
    __shared__ float ws[8];
    const int wave = threadIdx.x >> 5;
    if ((threadIdx.x & 31) == 0) ws[wave] = acc;
    __syncthreads();
    if (threadIdx.x == 0) {
        float t = 0.f;
        #pragma unroll
        for (int w = 0; w < 8; ++w) t += ws[w];
        if (r < NB) xsq[r] = t; else ysq[r - NB] = t;
    }
}

// ---------------------------------------------------------------------------
// Kernel 3: out[b][c] = 2 * (q_b . p_c) - xsq[b] - ysq[c]  (== -distance)
// One block per 16x16 output tile (8x4 tiles). 8 waves split K=16000 into
// 2000-wide slabs. Each wave keeps FOUR independent V_WMMA_F32_16X16X4_F32
// accumulator chains (k, k+4, k+8, k+12) so the XDL pipe isn't serialized on
// the D->C RAW dependency; partials reduced through LDS, epilogue fused.
//
// f32 16x16x4 fragment layout (ISA 7.12.2):
//   A: lane L -> row M = L&15, K pair = (L>>4)*2 .. +1   (2 VGPRs = v2f)
//   B: lane L -> col N = L&15, K pair = (L>>4)*2 .. +1   (B[k][n] = proto[n][k])
//   C/D: VGPR i -> (M = i + 8*(L>>4), N = L&15)
// ---------------------------------------------------------------------------
#define KWAVES 8
#define KPW (ND / KWAVES)   // 2000, multiple of 16

__global__ void __launch_bounds__(256, 1)
proto_dist_wmma_kernel(const float* __restrict__ query,
                       const float* __restrict__ proto,
                       const float* __restrict__ xsq,
                       const float* __restrict__ ysq,
                       float* __restrict__ out) {
    const int tileM = blockIdx.x >> 2;   // 0..7  (query rows / 16)
    const int tileN = blockIdx.x & 3;    // 0..3  (classes / 16)
    const int lane  = threadIdx.x & 31;
    const int wave  = threadIdx.x >> 5;

    const int rsel = lane & 15;          // M for A-frag, N for B-frag
    const int ksub = (lane >> 4) * 2;    // 0 or 2 within each K-quad

    const float* arow = query + (size_t)(tileM * 16 + rsel) * ND + ksub;
    const float* brow = proto + (size_t)(tileN * 16 + rsel) * ND + ksub;

    v8f c0 = {0.f, 0.f, 0.f, 0.f, 0.f, 0.f, 0.f, 0.f};
    v8f c1 = c0, c2 = c0, c3 = c0;
    const int k0 = wave * KPW;
    const int k1 = k0 + KPW;
    #pragma unroll 2
    for (int k = k0; k < k1; k += 16) {
        v2f a0 = *reinterpret_cast<const v2f*>(arow + k);
        v2f b0 = *reinterpret_cast<const v2f*>(brow + k);
        v2f a1 = *reinterpret_cast<const v2f*>(arow + k + 4);
        v2f b1 = *reinterpret_cast<const v2f*>(brow + k + 4);
        v2f a2 = *reinterpret_cast<const v2f*>(arow + k + 8);
        v2f b2 = *reinterpret_cast<const v2f*>(brow + k + 8);
        v2f a3 = *reinterpret_cast<const v2f*>(arow + k + 12);
        v2f b3 = *reinterpret_cast<const v2f*>(brow + k + 12);
        // (neg_a, A, neg_b, B, c_mod, C, reuse_a, reuse_b)
        c0 = __builtin_amdgcn_wmma_f32_16x16x4_f32(false, a0, false, b0,
                                                   (short)0, c0, false, false);
        c1 = __builtin_amdgcn_wmma_f32_16x16x4_f32(false, a1, false, b1,
                                                   (short)0, c1, false, false);
        c2 = __builtin_amdgcn_wmma_f32_16x16x4_f32(false, a2, false, b2,
                                                   (short)0, c2, false, false);
        c3 = __builtin_amdgcn_wmma_f32_16x16x4_f32(false, a3, false, b3,
                                                   (short)0, c3, false, false);
    }
    v8f c = (c0 + c1) + (c2 + c3);

    __shared__ float red[KWAVES][32][9];   // pad to 9 to dodge bank conflicts
    #pragma unroll
    for (int i = 0; i < 8; ++i) red[wave][lane][i] = c[i];
    __syncthreads();

    if (wave == 0) {
        const int n     = tileN * 16 + (lane & 15);
        const int mbase = tileM * 16 + 8 * (lane >> 4);
        const float yn  = ysq[n];
        #pragma unroll
        for (int i = 0; i < 8; ++i) {
            float t = red[0][lane][i];
            #pragma unroll
            for (int w = 1; w < KWAVES; ++w) t += red[w][lane][i];
            const int m = mbase + i;
            out[m * NC + n] = 2.0f * t - xsq[m] - yn;
        }
    }
}

// ---------------------------------------------------------------------------
extern "C" void kernel_launch(void* const* d_in, const int* in_sizes, int n_in,
                              void* d_out, int out_size, void* d_ws, size_t ws_size,
                              hipStream_t stream) {
    (void)in_sizes; (void)n_in; (void)out_size; (void)ws_size;
    const float* query   = (const float*)d_in[0];  // [128, 640, 5, 5]
    const float* support = (const float*)d_in[1];  // [64, 128, 640, 5, 5]
    float* out   = (float*)d_out;                  // [128, 64]

    float* proto = (float*)d_ws;                   // 64 * 16000 floats
    float* xsq   = proto + (size_t)NC * ND;        // 128 floats
    float* ysq   = xsq + NB;                       // 64 floats

    dim3 g1((ND / 4 + 255) / 256, NC);             // (16, 64)
    proto_mean_kernel<<<g1, 256, 0, stream>>>(support, proto);

    row_sumsq_kernel<<<NB + NC, 256, 0, stream>>>(query, proto, xsq, ysq);

    proto_dist_wmma_kernel<<<(NB / 16) * (NC / 16), 256, 0, stream>>>(
        query, proto, xsq, ysq, out);
}